// GraphAttentionLayer_38010460569729
// MI455X (gfx1250) — compile-verified
//
#include <hip/hip_runtime.h>

typedef __attribute__((ext_vector_type(2))) float v2f;
typedef __attribute__((ext_vector_type(8))) float v8f;

#define GAT_N 12288
#define IN_F  256
#define OUT_F 128
#define SLOPE 0.2f
#define CHUNK 256
#define LDS_PITCH (CHUNK + 4)   // 260 floats: row stride = 4 banks -> conflict-free A fetch

__device__ __forceinline__ v8f wmma_f32_16x16x4(v2f a, v2f b, v8f c) {
  // 8 args: (neg_a, A, neg_b, B, c_mod, C, reuse_a, reuse_b)
  return __builtin_amdgcn_wmma_f32_16x16x4_f32(false, a, false, b, (short)0, c,
                                               false, false);
}

// ---------------- Kernel 1: h_t = h @ W  (fp32 WMMA) ----------------
__global__ __launch_bounds__(256) void gat_ht_kernel(const float* __restrict__ h,
                                                     const float* __restrict__ Wm,
                                                     float* __restrict__ ht) {
  const int t    = threadIdx.x;
  const int lane = t & 31;
  const int half = lane >> 4;
  const int lr   = lane & 15;
  const int w    = t >> 5;
  const int m0   = blockIdx.x * 16;
  const int n0   = w * 16;
  v8f c = {};
  for (int k = 0; k < IN_F; k += 4) {
    const int kk = k + 2 * half;
    const float* ap = h + (size_t)(m0 + lr) * IN_F + kk;
    v2f av; av.x = ap[0]; av.y = ap[1];
    const float* bp = Wm + (size_t)kk * OUT_F + n0 + lr;
    v2f bv; bv.x = bp[0]; bv.y = bp[OUT_F];
    c = wmma_f32_16x16x4(av, bv, c);
  }
#pragma unroll
  for (int v = 0; v < 8; ++v)
    ht[(size_t)(m0 + v + 8 * half) * OUT_F + n0 + lr] = c[v];
}

// ---------------- Kernel 2: src/tgt = h_t @ a[:128], h_t @ a[128:] ----------------
__global__ __launch_bounds__(128) void gat_srctgt_kernel(const float* __restrict__ ht,
                                                         const float* __restrict__ a,
                                                         float* __restrict__ src,
                                                         float* __restrict__ tgt) {
  const int i = blockIdx.x, t = threadIdx.x;
  const float v = ht[(size_t)i * OUT_F + t];
  float s = v * a[t];
  float g = v * a[OUT_F + t];
#pragma unroll
  for (int off = 16; off > 0; off >>= 1) {
    s += __shfl_xor(s, off, 32);
    g += __shfl_xor(g, off, 32);
  }
  __shared__ float ss[4], gg[4];
  if ((t & 31) == 0) { ss[t >> 5] = s; gg[t >> 5] = g; }
  __syncthreads();
  if (t == 0) {
    src[i] = ss[0] + ss[1] + ss[2] + ss[3];
    tgt[i] = gg[0] + gg[1] + gg[2] + gg[3];
  }
}

// ---------------- Kernel 3: global max of tgt ----------------
__global__ __launch_bounds__(1024) void gat_tgtmax_kernel(const float* __restrict__ tgt,
                                                          float* __restrict__ tmax) {
  __shared__ float red[1024];
  const int t = threadIdx.x;
  float m = -3.4e38f;
  for (int c = t; c < GAT_N; c += 1024) m = fmaxf(m, tgt[c]);
  red[t] = m;
  __syncthreads();
  for (int s = 512; s > 0; s >>= 1) {
    if (t < s) red[t] = fmaxf(red[t], red[t + s]);
    __syncthreads();
  }
  if (t == 0) tmax[0] = red[0];
}

// ---------------- Kernel 4: fused mask + leakyrelu + softmax + (attn @ h_t) ----------------
// One block per 32-row stripe (two 16-row WMMA row-groups per wave, shared B
// operand). Pass 1 streams adj (NT) to get exp row-sums (stable via the
// global-tgt-max upper bound). Pass 2 recomputes p, writes normalized
// attention with NT stores, stages p in padded LDS and accumulates h_prime
// with v_wmma_f32_16x16x4_f32 (wave w owns output columns [16w, 16w+16)).
__global__ __launch_bounds__(256) void gat_attn_kernel(const int* __restrict__ adj,
                                                       const float* __restrict__ ht,
                                                       const float* __restrict__ src,
                                                       const float* __restrict__ tgt,
                                                       const float* __restrict__ tmaxp,
                                                       float* __restrict__ h_prime,
                                                       float* __restrict__ attn) {
  __shared__ float p_lds[32][LDS_PITCH];   // ~33 KB, bank-conflict-free
  __shared__ float rowsum[32];
  const int t    = threadIdx.x;
  const int r    = t >> 4;            // 0..15; thread scores rows r and r+16
  const int cl   = t & 15;
  const int row0 = blockIdx.x * 32;

  const float tmax = tmaxp[0];
  const float s0 = src[row0 + r];
  const float s1 = src[row0 + r + 16];
  float m0 = s0 + tmax; m0 = (m0 >= 0.f) ? m0 : SLOPE * m0;
  float m1 = s1 + tmax; m1 = (m1 >= 0.f) ? m1 : SLOPE * m1;
  const int* adj0 = adj + (size_t)(row0 + r) * GAT_N;
  const int* adj1 = adj + (size_t)(row0 + r + 16) * GAT_N;

  // Pass 1: unnormalized row sums of exp
  float acc0 = 0.f, acc1 = 0.f;
  for (int c0 = 0; c0 < GAT_N; c0 += CHUNK) {
#pragma unroll
    for (int j = 0; j < CHUNK / 16; ++j) {
      const int c  = c0 + cl + 16 * j;
      const float tv = tgt[c];
      const int a0 = __builtin_nontemporal_load(adj0 + c);
      const int a1 = __builtin_nontemporal_load(adj1 + c);
      float e0 = s0 + tv; e0 = (e0 >= 0.f) ? e0 : SLOPE * e0;
      float e1 = s1 + tv; e1 = (e1 >= 0.f) ? e1 : SLOPE * e1;
      acc0 += (a0 > 0) ? __expf(e0 - m0) : 0.f;
      acc1 += (a1 > 0) ? __expf(e1 - m1) : 0.f;
    }
  }
#pragma unroll
  for (int off = 8; off > 0; off >>= 1) {
    acc0 += __shfl_xor(acc0, off, 32);
    acc1 += __shfl_xor(acc1, off, 32);
  }
  if (cl == 0) { rowsum[r] = acc0; rowsum[r + 16] = acc1; }
  __syncthreads();
  float inv0 = rowsum[r];      inv0 = (inv0 > 0.f) ? 1.f / inv0 : 0.f;
  float inv1 = rowsum[r + 16]; inv1 = (inv1 > 0.f) ? 1.f / inv1 : 0.f;

  // WMMA lane mapping
  const int lane = t & 31;
  const int half = lane >> 4;
  const int lr   = lane & 15;
  const int w    = t >> 5;
  const int n0   = w * 16;
  v8f ca = {};   // rows row0 .. row0+15
  v8f cb = {};   // rows row0+16 .. row0+31
  float* at0 = attn + (size_t)(row0 + r) * GAT_N;
  float* at1 = attn + (size_t)(row0 + r + 16) * GAT_N;

  for (int c0 = 0; c0 < GAT_N; c0 += CHUNK) {
    // Phase A: normalized attention tile -> global (NT) + LDS
#pragma unroll
    for (int j = 0; j < CHUNK / 16; ++j) {
      const int clocal = cl + 16 * j;
      const int c      = c0 + clocal;
      const float tv   = tgt[c];
      const int a0 = __builtin_nontemporal_load(adj0 + c);
      const int a1 = __builtin_nontemporal_load(adj1 + c);
      float e0 = s0 + tv; e0 = (e0 >= 0.f) ? e0 : SLOPE * e0;
      float e1 = s1 + tv; e1 = (e1 >= 0.f) ? e1 : SLOPE * e1;
      const float p0 = (a0 > 0) ? __expf(e0 - m0) * inv0 : 0.f;
      const float p1 = (a1 > 0) ? __expf(e1 - m1) * inv1 : 0.f;
      p_lds[r][clocal]      = p0;
      p_lds[r + 16][clocal] = p1;
      __builtin_nontemporal_store(p0, at0 + c);
      __builtin_nontemporal_store(p1, at1 + c);
    }
    __syncthreads();
    // Phase B: two 16x16 row-group accumulations sharing the B operand
    for (int k = 0; k < CHUNK; k += 4) {
      const int kk = k + 2 * half;
      v2f a0v; a0v.x = p_lds[lr][kk];      a0v.y = p_lds[lr][kk + 1];
      v2f a1v; a1v.x = p_lds[lr + 16][kk]; a1v.y = p_lds[lr + 16][kk + 1];
      const float* bp = ht + (size_t)(c0 + kk) * OUT_F + n0 + lr;
      v2f bv; bv.x = bp[0]; bv.y = bp[OUT_F];
      ca = wmma_f32_16x16x4(a0v, bv, ca);
      cb = wmma_f32_16x16x4(a1v, bv, cb);
    }
    __syncthreads();
  }
#pragma unroll
  for (int v = 0; v < 8; ++v)
    h_prime[(size_t)(row0 + v + 8 * half) * OUT_F + n0 + lr] = ca[v];
#pragma unroll
  for (int v = 0; v < 8; ++v)
    h_prime[(size_t)(row0 + 16 + v + 8 * half) * OUT_F + n0 + lr] = cb[v];
}

extern "C" void kernel_launch(void* const* d_in, const int* in_sizes, int n_in,
                              void* d_out, int out_size, void* d_ws, size_t ws_size,
                              hipStream_t stream) {
  const float* h   = (const float*)d_in[0];   // [N, IN_F] f32
  const int*   adj = (const int*)d_in[1];     // [N, N] i32
  const float* Wm  = (const float*)d_in[2];   // [IN_F, OUT_F] f32
  const float* a   = (const float*)d_in[3];   // [2*OUT_F, 1] f32

  float* ht   = (float*)d_ws;                     // N*OUT_F
  float* src  = ht  + (size_t)GAT_N * OUT_F;      // N
  float* tgt  = src + GAT_N;                      // N
  float* tmax = tgt + GAT_N;                      // 1

  float* h_prime = (float*)d_out;                      // [N, OUT_F]
  float* attn    = h_prime + (size_t)GAT_N * OUT_F;    // [N, N]

  gat_ht_kernel<<<GAT_N / 16, 256, 0, stream>>>(h, Wm, ht);
  gat_srctgt_kernel<<<GAT_N, 128, 0, stream>>>(ht, a, src, tgt);
  gat_tgtmax_kernel<<<1, 1024, 0, stream>>>(tgt, tmax);
  gat_attn_kernel<<<GAT_N / 32, 256, 0, stream>>>(adj, ht, src, tgt, tmax,
                                                  h_prime, attn);
}